// Appnp_64433099375271
// MI455X (gfx1250) — compile-verified
//
#include <hip/hip_runtime.h>
#include <hip/hip_bf16.h>

// APPNP for MI455X (gfx1250).
//
// Roofline reasoning:
//  * GEMMs are bandwidth-bound (3.3 GFLOP vs 102 MB of activations) -> use
//    V_WMMA_F32_16X16X4_F32 (full fp32, matches reference numerics, matrix pipe).
//  * Propagation working set (~140 MB: h/x0/agg + int32 edges + weights) fits the
//    192 MB L2, so the 845 MB/iter gather + 845 MB/iter scatter-add run L2-resident
//    and atomics resolve at L2 atomic units. Keep fp32 (reference is fp32).
//  * Scatter uses hardware global_atomic_add_f32 (no-return form, STOREcnt) via
//    inline asm so we never fall back to a CAS loop.

#define NFEAT  256
#define HDIM   64
#define NCLASS 64
#define KPROP  4
#define ALPHA  0.1f

typedef float v2f __attribute__((ext_vector_type(2)));
typedef float v8f __attribute__((ext_vector_type(8)));

// ---------------------------------------------------------------------------
// Hardware fp32 global atomic add (no return -> tracked by STOREcnt; s_endpgm
// performs an implicit wait-idle so no explicit s_wait_storecnt is needed).
// ---------------------------------------------------------------------------
__device__ __forceinline__ void gatomic_add_f32(float* p, float v) {
    asm volatile("global_atomic_add_f32 %0, %1, off" : : "v"(p), "v"(v) : "memory");
}

// ---------------------------------------------------------------------------
// Degree / normalization kernels
// ---------------------------------------------------------------------------
__global__ void init_deg_kernel(float* __restrict__ deg, int n) {
    int i = blockIdx.x * blockDim.x + threadIdx.x;
    if (i < n) deg[i] = 1.0f;                 // self-loop contributes 1 to degree
}

__global__ void deg_kernel(const int* __restrict__ dst, float* __restrict__ deg, int E) {
    int e = blockIdx.x * blockDim.x + threadIdx.x;
    if (e < E) gatomic_add_f32(deg + dst[e], 1.0f);
}

__global__ void rsqrt_kernel(float* __restrict__ dinv, int n) {
    int i = blockIdx.x * blockDim.x + threadIdx.x;
    if (i < n) {
        float d = dinv[i];                    // deg >= 1 always (self loop)
        dinv[i] = 1.0f / sqrtf(d);
    }
}

__global__ void weight_kernel(const int* __restrict__ src, const int* __restrict__ dst,
                              const float* __restrict__ dinv, float* __restrict__ w, int E) {
    int e = blockIdx.x * blockDim.x + threadIdx.x;
    if (e < E) w[e] = dinv[src[e]] * dinv[dst[e]];
}

// ---------------------------------------------------------------------------
// WMMA fp32 GEMM:  C[nRows,64] = act(A[nRows,KDIM]) @ W[64,KDIM]^T + bias
// Block = 128 threads = 4 waves; wave w owns N-tile n0 = 16*w, block owns one
// M-tile of 16 rows.  K consumed 4 at a time via V_WMMA_F32_16X16X4_F32.
//
// VGPR layouts (ISA 7.12.2, 32-bit):
//   A 16x4 : lanes 0-15 -> M=lane, {v0,v1} = K{0,1}; lanes 16-31 -> K{2,3}
//   B 4x16 : lanes 0-15 -> N=lane, {v0,v1} = K{0,1}; lanes 16-31 -> K{2,3}
//   C/D    : vgpr r, lanes 0-15 -> M=r, N=lane; lanes 16-31 -> M=8+r, N=lane-16
// EXEC is all-1s: no divergence anywhere in this kernel (nRows % 16 == 0).
// ---------------------------------------------------------------------------
template <int KDIM, bool RELU>
__global__ void gemm_wmma_kernel(const float* __restrict__ A,
                                 const float* __restrict__ W,
                                 const float* __restrict__ bias,
                                 float* __restrict__ C,
                                 int nRows) {
    const int lane = threadIdx.x & 31;
    const int wave = threadIdx.x >> 5;        // 0..3 -> N tile
    const int row0 = blockIdx.x * 16;
    const int n0   = wave * 16;
    const int half = lane >> 4;               // 0: K{0,1}, 1: K{2,3}
    const int l    = lane & 15;

    const float* Arow = A + (size_t)(row0 + l) * KDIM;
    const float* Wrow = W + (size_t)(n0 + l) * KDIM;

    v8f acc = {};
#pragma unroll 4
    for (int k = 0; k < KDIM; k += 4) {
        const int kk = k + 2 * half;
        float a0 = Arow[kk];
        float a1 = Arow[kk + 1];
        if (RELU) { a0 = fmaxf(a0, 0.0f); a1 = fmaxf(a1, 0.0f); }
        v2f a = {a0, a1};
        v2f b = {Wrow[kk], Wrow[kk + 1]};
        acc = __builtin_amdgcn_wmma_f32_16x16x4_f32(
            /*neg_a=*/false, a, /*neg_b=*/false, b,
            /*c_mod=*/(short)0, acc, /*reuse_a=*/false, /*reuse_b=*/false);
    }

    const int   n  = n0 + l;
    const float bn = bias[n];
    float* Cb = C + (size_t)(row0 + 8 * half) * HDIM + n;   // HDIM == NCLASS == 64
#pragma unroll
    for (int r = 0; r < 8; ++r)
        Cb[(size_t)r * HDIM] = acc[r] + bn;
}

// ---------------------------------------------------------------------------
// Propagation kernels
// ---------------------------------------------------------------------------
__global__ void zero_kernel(float4* __restrict__ p, int n4) {
    int i = blockIdx.x * blockDim.x + threadIdx.x;
    if (i < n4) p[i] = make_float4(0.f, 0.f, 0.f, 0.f);
}

// One edge handled by 16 consecutive lanes (4 features each): the h[src] gather
// is one fully-coalesced 256B read per edge; scatter = 4 hw fp32 atomics/lane.
__global__ void scatter_kernel(const int* __restrict__ src, const int* __restrict__ dst,
                               const float* __restrict__ w, const float* __restrict__ h,
                               float* __restrict__ agg, int E) {
    int t = blockIdx.x * blockDim.x + threadIdx.x;   // E*16 threads total
    int e = t >> 4;
    if (e >= E) return;
    int chunk = t & 15;

    int   s  = src[e];
    int   d  = dst[e];
    float we = w[e];

    const float4* hs = (const float4*)(h + (size_t)s * HDIM);
    float4 v = hs[chunk];

    float* out = agg + (size_t)d * HDIM + chunk * 4;
    gatomic_add_f32(out + 0, we * v.x);
    gatomic_add_f32(out + 1, we * v.y);
    gatomic_add_f32(out + 2, we * v.z);
    gatomic_add_f32(out + 3, we * v.w);
}

// h_next = (1-a) * (agg + dinv[i]^2 * h_cur) + a * x0    (self-loop folded in);
// operates in place on agg.
__global__ void update_kernel(float4* __restrict__ hnext_agg,
                              const float4* __restrict__ hcur,
                              const float4* __restrict__ x0,
                              const float* __restrict__ dinv, int n4) {
    int i = blockIdx.x * blockDim.x + threadIdx.x;   // N*16 float4 chunks
    if (i >= n4) return;
    int node = i >> 4;                                // 16 float4 per node (64 feats)
    float sw = dinv[node];
    sw *= sw;

    float4 a = hnext_agg[i];
    float4 h = hcur[i];
    float4 x = x0[i];
    float4 r;
    r.x = (1.0f - ALPHA) * (a.x + sw * h.x) + ALPHA * x.x;
    r.y = (1.0f - ALPHA) * (a.y + sw * h.y) + ALPHA * x.y;
    r.z = (1.0f - ALPHA) * (a.z + sw * h.z) + ALPHA * x.z;
    r.w = (1.0f - ALPHA) * (a.w + sw * h.w) + ALPHA * x.w;
    hnext_agg[i] = r;
}

// ---------------------------------------------------------------------------
// Launch
// ---------------------------------------------------------------------------
extern "C" void kernel_launch(void* const* d_in, const int* in_sizes, int n_in,
                              void* d_out, int out_size, void* d_ws, size_t ws_size,
                              hipStream_t stream) {
    const float* x     = (const float*)d_in[0];   // [N, 256]
    const int*   ei    = (const int*)d_in[1];     // [2, E] int32 (JAX default x32)
    const float* W_in  = (const float*)d_in[2];   // [64, 256]
    const float* b_in  = (const float*)d_in[3];   // [64]
    const float* W_out = (const float*)d_in[4];   // [64, 64]
    const float* b_out = (const float*)d_in[5];   // [64]
    float*       out   = (float*)d_out;           // [N, 64]

    const int N = in_sizes[0] / NFEAT;            // 100000 (divisible by 16)
    const int E = in_sizes[1] / 2;                // 3200000
    const int* src = ei;
    const int* dst = ei + E;

    // Workspace layout (floats):  dinv[N] | w[E] | x0[N*64] | buf0[N*64] | buf1[N*64]
    // Total ~90 MB; every buffer we read is written first (no reliance on poison).
    float* dinv = (float*)d_ws;
    float* w    = dinv + N;
    float* x0   = w + E;
    float* buf0 = x0 + (size_t)N * HDIM;
    float* buf1 = buf0 + (size_t)N * HDIM;

    const int B = 256;

    // GCN normalization: deg (with self-loop), dinv = deg^-1/2, w = dinv[s]*dinv[d]
    init_deg_kernel<<<(N + B - 1) / B, B, 0, stream>>>(dinv, N);
    deg_kernel<<<(E + B - 1) / B, B, 0, stream>>>(dst, dinv, E);
    rsqrt_kernel<<<(N + B - 1) / B, B, 0, stream>>>(dinv, N);
    weight_kernel<<<(E + B - 1) / B, B, 0, stream>>>(src, dst, dinv, w, E);

    // h = x @ W_in^T + b_in  (WMMA fp32)
    gemm_wmma_kernel<NFEAT, false><<<N / 16, 128, 0, stream>>>(x, W_in, b_in, x0, N);

    // K APPNP propagation steps with ping-pong buffers (self-loop folded in update)
    const int n4 = N * (HDIM / 4);
    const float* cur = x0;
    float* bufs[2] = {buf0, buf1};
    for (int k = 0; k < KPROP; ++k) {
        float* agg = bufs[k & 1];
        zero_kernel<<<(n4 + B - 1) / B, B, 0, stream>>>((float4*)agg, n4);
        scatter_kernel<<<((size_t)E * 16 + B - 1) / B, B, 0, stream>>>(src, dst, w, cur, agg, E);
        update_kernel<<<(n4 + B - 1) / B, B, 0, stream>>>((float4*)agg, (const float4*)cur,
                                                          (const float4*)x0, dinv, n4);
        cur = agg;
    }

    // out = relu(h) @ W_out^T + b_out  (WMMA fp32, ReLU fused into A load)
    gemm_wmma_kernel<HDIM, true><<<N / 16, 128, 0, stream>>>(cur, W_out, b_out, out, N);
}